// GaussianSplatter_22634477650078
// MI455X (gfx1250) — compile-verified
//
#include <hip/hip_runtime.h>
#include <hip/hip_bf16.h>

typedef float v2f __attribute__((ext_vector_type(2)));
typedef float v8f __attribute__((ext_vector_type(8)));

// ---------------------------------------------------------------------------
// Pass 1: reduce logits [16,100,126,126] -> P[c*16+b][49]
// S[c][p] = sum over (b-slice) of logits at in-patch position p
// Block = (c, b). Thread x = column w (fixed j = w%7). 7 regs per thread
// accumulate over unrolled i = h%7, then LDS reduction over the 18 patch cols.
// ---------------------------------------------------------------------------
__global__ void __launch_bounds__(128)
k1_reduce_logits(const float* __restrict__ logits, float* __restrict__ P) {
    const int c = blockIdx.x;   // 0..99
    const int b = blockIdx.y;   // 0..15
    const int w = threadIdx.x;  // 0..127 (126 active for loads)
    __shared__ float lds[7 * 128];

    float acc[7];
#pragma unroll
    for (int i = 0; i < 7; ++i) acc[i] = 0.f;

    const float* base = logits + ((size_t)(b * 100 + c)) * 126 * 126;
    if (w < 126) {
        for (int pr = 0; pr < 18; ++pr) {
#pragma unroll
            for (int i = 0; i < 7; ++i) {
                acc[i] += base[(pr * 7 + i) * 126 + w];
            }
        }
    }
#pragma unroll
    for (int i = 0; i < 7; ++i) lds[i * 128 + w] = (w < 126) ? acc[i] : 0.f;
    __syncthreads();

    if (threadIdx.x < 49) {
        const int i = threadIdx.x / 7, j = threadIdx.x % 7;
        float s = 0.f;
#pragma unroll
        for (int kk = 0; kk < 18; ++kk) s += lds[i * 128 + j + 7 * kk];
        P[(c * 16 + b) * 49 + threadIdx.x] = s;
    }
}

// ---------------------------------------------------------------------------
// Pass 2: single block. Reduce partials -> weighted gaussian params,
// build padded kernels, bilinear shift, write Kg[p][q] = wop[p]*kt[p][q].
// ---------------------------------------------------------------------------
__global__ void __launch_bounds__(64)
k2_params(const float* __restrict__ P,
          const float* __restrict__ sx, const float* __restrict__ sy,
          const float* __restrict__ op, const float* __restrict__ rho,
          float* __restrict__ Kg /* 49*49 */) {
    __shared__ float pad[49][49];  // padded 7x7 kernel per point p
    const int p = threadIdx.x;     // 49 active of 64
    float wop = 0.f;

    if (p < 49) {
        float wsx = 0.f, wsy = 0.f, wrh = 0.f;
        for (int c = 0; c < 100; ++c) {
            float s = 0.f;
            for (int b = 0; b < 16; ++b) s += P[(c * 16 + b) * 49 + p];
            wsx += sx[c] * s;
            wsy += sy[c] * s;
            wop += op[c] * s;
            wrh += rho[c] * s;
        }
        const float inv = 1.0f / 5184.0f;  // mean over BL
        wsx *= inv; wsy *= inv; wop *= inv; wrh *= inv;

        const float a  = wsx * wsx + 1e-5f;
        const float d  = wsy * wsy + 1e-5f;
        const float bb = wrh * wsx * wsy;
        const float det = a * d - bb * bb;
        const float ninvdet2 = -0.5f / det;

        // padded kernel: kern/max == exp(z) (quadratic form is PD, max at center)
#pragma unroll
        for (int y = 0; y < 7; ++y) {
#pragma unroll
            for (int x = 0; x < 7; ++x) {
                float v = 0.f;
                if (y >= 1 && y <= 5 && x >= 1 && x <= 5) {
                    const float gx = -5.f + 2.5f * (float)(y - 1);
                    const float gy = -5.f + 2.5f * (float)(x - 1);
                    const float z = (d * gx * gx - 2.f * bb * gx * gy + a * gy * gy) * ninvdet2;
                    v = expf(z);
                }
                pad[p][y * 7 + x] = v;
            }
        }
    }
    __syncthreads();

    if (p < 49) {
        const float ty = (1.f - 2.f * (float)(p / 7) / 7.f) * 3.f;
        const float tx = (1.f - 2.f * (float)(p % 7) / 7.f) * 3.f;
#pragma unroll
        for (int i = 0; i < 7; ++i) {
            const float yq = (float)i + ty;
            const float y0f = floorf(yq);
            const float fy = yq - y0f;
            const int y0 = (int)y0f;
#pragma unroll
            for (int j = 0; j < 7; ++j) {
                const float xq = (float)j + tx;
                const float x0f = floorf(xq);
                const float fx = xq - x0f;
                const int x0 = (int)x0f;
                float acc = 0.f;
#pragma unroll
                for (int dy = 0; dy < 2; ++dy) {
                    const int yi = y0 + dy;
                    const float wy = dy ? fy : (1.f - fy);
                    if (yi < 0 || yi >= 7) continue;
#pragma unroll
                    for (int dx = 0; dx < 2; ++dx) {
                        const int xi = x0 + dx;
                        const float wx = dx ? fx : (1.f - fx);
                        if (xi < 0 || xi >= 7) continue;
                        acc += wy * wx * pad[p][yi * 7 + xi];
                    }
                }
                Kg[p * 49 + i * 7 + j] = wop * acc;
            }
        }
    }
}

// ---------------------------------------------------------------------------
// Pass 3: splat GEMM via V_WMMA_F32_16X16X4_F32.
//   out[m][q] = sum_p X[m][p] * Kg[p][q],  M=41472 rows (b,c,pr,pc), K=N=49.
// K padded to 52 (13 steps), N padded to 64 (4 n-tiles). 1 wave = 16-row tile.
// A layout (16x4 f32): lane = m + 16*(k>>1), vgpr = k&1.
// B layout (4x16 f32): lane = n + 16*(k>>1), vgpr = k&1 (from LDS).
// D layout (16x16 f32): vgpr r -> M = r + 8*(lane>>4), N = lane&15.
// ---------------------------------------------------------------------------
__global__ void __launch_bounds__(128)
k3_splat(const float* __restrict__ feat, const float* __restrict__ Kg,
         float* __restrict__ out) {
    __shared__ float KB[52 * 64];  // [k][n], zero padded
    for (int idx = threadIdx.x; idx < 52 * 64; idx += 128) {
        const int k = idx >> 6, n = idx & 63;
        KB[idx] = (k < 49 && n < 49) ? Kg[k * 49 + n] : 0.f;
    }
    __syncthreads();

    const int lane = threadIdx.x & 31;
    const int hi = lane >> 4;  // selects K pair {0,1} vs {2,3} within each step
    const int lo = lane & 15;
    const int wid = blockIdx.x * 4 + (threadIdx.x >> 5);
    const int m0 = wid * 16;

    // A row for this lane: m = m0 + lo -> (b, c, pr, pc)
    {
        // nothing: computed below
    }
    int m = m0 + lo;
    const int pc = m % 18; int t = m / 18;
    const int pr = t % 18; t /= 18;
    const int ch = t % 8;  const int bI = t / 8;
    const float* rowbase =
        feat + (((size_t)(bI * 8 + ch) * 126 + pr * 7) * 126 + pc * 7);

    v8f acc[4] = {v8f{}, v8f{}, v8f{}, v8f{}};

#pragma unroll
    for (int s = 0; s < 13; ++s) {
        const int k0 = 4 * s + 2 * hi;
        v2f a;
        {
            const int p0 = k0, p1 = k0 + 1;
            float a0 = 0.f, a1 = 0.f;
            if (p0 < 49) a0 = rowbase[(p0 / 7) * 126 + (p0 % 7)];
            if (p1 < 49) a1 = rowbase[(p1 / 7) * 126 + (p1 % 7)];
            a.x = a0;  // vgpr0 = even k of the pair
            a.y = a1;  // vgpr1 = odd k
        }
#pragma unroll
        for (int nt = 0; nt < 4; ++nt) {
            v2f bf;
            bf.x = KB[k0 * 64 + nt * 16 + lo];
            bf.y = KB[(k0 + 1) * 64 + nt * 16 + lo];
            acc[nt] = __builtin_amdgcn_wmma_f32_16x16x4_f32(
                /*neg_a=*/false, a, /*neg_b=*/false, bf,
                /*c_mod=*/(short)0, acc[nt],
                /*reuse_a=*/false, /*reuse_b=*/false);
        }
    }

    // Store D: vgpr r -> row m0 + r + 8*hi, col n = nt*16 + lo (mask n<49)
#pragma unroll
    for (int r = 0; r < 8; ++r) {
        const int mm = m0 + r + 8 * hi;
        const int pc2 = mm % 18; int tt = mm / 18;
        const int pr2 = tt % 18; tt /= 18;
        const int c2 = tt % 8;   const int b2 = tt / 8;
        float* ob = out + (((size_t)(b2 * 8 + c2) * 126 + pr2 * 7) * 126 + pc2 * 7);
#pragma unroll
        for (int nt = 0; nt < 4; ++nt) {
            const int n = nt * 16 + lo;
            if (n < 49) {
                ob[(n / 7) * 126 + (n % 7)] = acc[nt][r];
            }
        }
    }
}

// ---------------------------------------------------------------------------
extern "C" void kernel_launch(void* const* d_in, const int* in_sizes, int n_in,
                              void* d_out, int out_size, void* d_ws, size_t ws_size,
                              hipStream_t stream) {
    const float* feat   = (const float*)d_in[0];  // [16,8,126,126]
    const float* logits = (const float*)d_in[1];  // [16,100,126,126]
    const float* sx     = (const float*)d_in[2];  // [100]
    const float* sy     = (const float*)d_in[3];  // [100]
    const float* op     = (const float*)d_in[4];  // [100]
    const float* rho    = (const float*)d_in[5];  // [100]
    float* out = (float*)d_out;                   // [16,8,126,126]

    float* P  = (float*)d_ws;       // 1600*49 floats = 78400
    float* Kg = P + 1600 * 49;      // 49*49 floats

    dim3 g1(100, 16);
    k1_reduce_logits<<<g1, 128, 0, stream>>>(logits, P);
    k2_params<<<1, 64, 0, stream>>>(P, sx, sy, op, rho, Kg);
    k3_splat<<<648, 128, 0, stream>>>(feat, Kg, out);  // 648*4 waves = 2592 M-tiles
}